// Encoder_DGI_19928648253625
// MI455X (gfx1250) — compile-verified
//
#include <hip/hip_runtime.h>
#include <hip/hip_bf16.h>

#define NFEAT 256
#define NHID  128

typedef float v2f __attribute__((ext_vector_type(2)));
typedef float v8f __attribute__((ext_vector_type(8)));

__device__ __forceinline__ void atomAddF(float* p, float v) {
    // maps to global_atomic_add_f32 on gfx1250 (no CAS loop)
    unsafeAtomicAdd(p, v);
}

// ---------------- Kernel 1: spectral norm + W_sn = W / sigma ----------------
// Single block, 256 threads. W is [256,128] row-major. sigma == ||W @ v||.
__global__ __launch_bounds__(256) void k_specnorm(const float* __restrict__ W,
                                                  const float* __restrict__ u,
                                                  float* __restrict__ wsn) {
    __shared__ float v_sh[NHID];
    __shared__ float red[256];
    const int tid = threadIdx.x;

    // v_raw = W^T u   (thread j < 128 computes column j, coalesced over tid)
    float vraw = 0.f;
    if (tid < NHID) {
        for (int i = 0; i < NFEAT; ++i) vraw += W[i * NHID + tid] * u[i];
    }
    red[tid] = (tid < NHID) ? vraw * vraw : 0.f;
    __syncthreads();
    for (int s = 128; s > 0; s >>= 1) {
        if (tid < s) red[tid] += red[tid + s];
        __syncthreads();
    }
    const float vnorm = sqrtf(red[0]) + 1e-12f;   // read before red is reused
    __syncthreads();
    if (tid < NHID) v_sh[tid] = vraw / vnorm;
    __syncthreads();

    // wv = W @ v  (thread i computes row i); sigma = ||wv||^2 / (||wv|| + eps)
    float wv = 0.f;
    for (int j = 0; j < NHID; ++j) wv += W[tid * NHID + j] * v_sh[j];
    red[tid] = wv * wv;
    __syncthreads();
    for (int s = 128; s > 0; s >>= 1) {
        if (tid < s) red[tid] += red[tid + s];
        __syncthreads();
    }
    const float s2 = red[0];
    const float sigma = s2 / (sqrtf(s2) + 1e-12f);
    const float inv = 1.0f / sigma;

    for (int idx = tid; idx < NFEAT * NHID; idx += 256) wsn[idx] = W[idx] * inv;
}

// ---------------- Kernel 2: zero a float buffer ----------------
__global__ void k_zero(float* __restrict__ p, int n) {
    int i = blockIdx.x * blockDim.x + threadIdx.x;
    if (i < n) p[i] = 0.f;
}

// ---------------- Kernel 3: degree over targets (col) ----------------
__global__ void k_degree(const int* __restrict__ ei, int E, float* __restrict__ deg) {
    int i = blockIdx.x * blockDim.x + threadIdx.x;
    if (i < E) atomAddF(&deg[ei[E + i]], 1.0f);   // col = ei[1][i]
}

// ---------------- Kernel 4: dinv = rsqrt(deg + 1) (self loop) ----------------
__global__ void k_dinv(float* __restrict__ deg, int N) {
    int i = blockIdx.x * blockDim.x + threadIdx.x;
    if (i < N) deg[i] = rsqrtf(deg[i] + 1.0f);
}

// ---------------- Kernel 5: xw = x @ W_sn via V_WMMA_F32_16X16X4_F32 --------
// 256 threads = 8 wave32s per block. Block owns 16 rows of x (K=256 staged in
// LDS); wave w owns output tile [16 x 16] at n0 = 16*w (NHID = 128 = 8*16).
__global__ __launch_bounds__(256) void k_gemm_wmma(const float* __restrict__ x,
                                                   const float* __restrict__ wsn,
                                                   float* __restrict__ xw) {
    __shared__ float xs[16][260];   // stride 260 -> conflict-free column reads
    const int tid = threadIdx.x;
    const int m0 = blockIdx.x * 16;

    // cooperative load of x[m0..m0+15][0..255] (4096 floats, float4 per thread x4)
    #pragma unroll
    for (int i = 0; i < 4; ++i) {
        const int r = i * 4 + (tid >> 6);
        const int c = (tid & 63) * 4;
        const float4 v = *reinterpret_cast<const float4*>(
            x + (size_t)(m0 + r) * NFEAT + c);
        xs[r][c + 0] = v.x; xs[r][c + 1] = v.y;
        xs[r][c + 2] = v.z; xs[r][c + 3] = v.w;
    }
    __syncthreads();

    const int wave = tid >> 5;
    const int lane = tid & 31;
    const int hi   = lane >> 4;     // lane half (0: lanes 0-15, 1: lanes 16-31)
    const int l    = lane & 15;
    const int n0   = wave * 16;

    v8f acc = {};
    const float* wp = wsn + n0 + l;     // column n0+l of W_sn

    #pragma unroll 8
    for (int k = 0; k < NFEAT; k += 4) {
        const int ka = k + 2 * hi;
        v2f a, b;
        // A 16x4 f32 layout: VGPR0 = K0|K2, VGPR1 = K1|K3 per lane half
        a.x = xs[l][ka];
        a.y = xs[l][ka + 1];
        // B 4x16 f32 layout: rows striped across lanes, analogous halves
        b.x = wp[(size_t)ka * NHID];
        b.y = wp[(size_t)(ka + 1) * NHID];
        acc = __builtin_amdgcn_wmma_f32_16x16x4_f32(
            /*neg_a=*/false, a, /*neg_b=*/false, b,
            /*c_mod=*/(short)0, acc, /*reuse_a=*/false, /*reuse_b=*/false);
    }

    // D layout: VGPR r -> M = r (+8 for lane half 1), N = n0 + l
    float* op = xw + (size_t)(m0 + 8 * hi) * NHID + n0 + l;
    #pragma unroll
    for (int r = 0; r < 8; ++r) op[(size_t)r * NHID] = acc[r];
}

// ---------------- Kernel 6: edge scatter-add (one wave per edge) ------------
__global__ __launch_bounds__(256) void k_scatter(const int* __restrict__ ei, int E,
                                                 const float* __restrict__ xw,
                                                 const float* __restrict__ dinv,
                                                 float* __restrict__ out) {
    const int e = blockIdx.x * 8 + (threadIdx.x >> 5);
    if (e >= E) return;
    const int lane = threadIdx.x & 31;
    const int r = ei[e];        // source
    const int c = ei[E + e];    // target
    const float w = dinv[r] * dinv[c];
    const float4 v = reinterpret_cast<const float4*>(xw + (size_t)r * NHID)[lane];
    float* dst = out + (size_t)c * NHID + lane * 4;
    atomAddF(dst + 0, w * v.x);
    atomAddF(dst + 1, w * v.y);
    atomAddF(dst + 2, w * v.z);
    atomAddF(dst + 3, w * v.w);
}

// ---------------- Kernel 7: self-loop + bias + PReLU ------------------------
__global__ void k_finalize(float* __restrict__ out, const float* __restrict__ xw,
                           const float* __restrict__ dinv,
                           const float* __restrict__ bias,
                           const float* __restrict__ prelu_a, int total) {
    int idx = blockIdx.x * blockDim.x + threadIdx.x;
    if (idx >= total) return;
    const int i = idx >> 7;         // node
    const int j = idx & (NHID - 1); // feature
    const float di = dinv[i];
    float v = out[idx] + di * di * xw[idx] + bias[j];
    const float a = prelu_a[0];
    out[idx] = v > 0.f ? v : a * v;
}

extern "C" void kernel_launch(void* const* d_in, const int* in_sizes, int n_in,
                              void* d_out, int out_size, void* d_ws, size_t ws_size,
                              hipStream_t stream) {
    const float* x       = (const float*)d_in[0];
    const int*   ei      = (const int*)  d_in[1];
    const float* W       = (const float*)d_in[2];
    const float* bias    = (const float*)d_in[3];
    const float* prelu_a = (const float*)d_in[4];
    const float* u       = (const float*)d_in[5];
    float* out = (float*)d_out;

    const int N = in_sizes[0] / NFEAT;   // 100000
    const int E = in_sizes[1] / 2;       // 1600000

    float* ws   = (float*)d_ws;
    float* wsn  = ws;                    // [NFEAT*NHID]
    float* dinv = ws + NFEAT * NHID;     // [N] (deg first, then dinv in-place)
    float* xw   = dinv + N;              // [N*NHID] (offset stays 16B aligned)

    const int total = N * NHID;

    // 1) sigma + W_sn
    k_specnorm<<<1, 256, 0, stream>>>(W, u, wsn);
    // 2) zero deg and out
    k_zero<<<(N + 255) / 256, 256, 0, stream>>>(dinv, N);
    k_zero<<<(total + 255) / 256, 256, 0, stream>>>(out, total);
    // 3) degree atomics, 4) dinv
    k_degree<<<(E + 255) / 256, 256, 0, stream>>>(ei, E, dinv);
    k_dinv<<<(N + 255) / 256, 256, 0, stream>>>(dinv, N);
    // 5) WMMA GEMM xw = x @ W_sn  (N divisible by 16)
    k_gemm_wmma<<<N / 16, 256, 0, stream>>>(x, wsn, xw);
    // 6) edge scatter
    k_scatter<<<(E + 7) / 8, 256, 0, stream>>>(ei, E, xw, dinv, out);
    // 7) self-loop + bias + PReLU
    k_finalize<<<(total + 255) / 256, 256, 0, stream>>>(out, xw, dinv, bias,
                                                        prelu_a, total);
}